// LoCoScaleAdaptive_83245056131239
// MI455X (gfx1250) — compile-verified
//
#include <hip/hip_runtime.h>
#include <math.h>

// ---------------------------------------------------------------------------
// LoCoScaleAdaptive on MI455X (gfx1250, wave32, WMMA).
// Sizes: B=8 N=2048 I=64 H=128 O=8 L=3 K=24, ITERS=3 TAU=0.1 LAM=0.5 KNN=5
// All GEMM-shaped work uses V_WMMA_F32_16X16X4_F32 (full f32 precision; the
// TAU=0.1-sharpened softmax makes f16 GEMM error risky, and the problem is
// L2-resident so precision costs ~nothing at 23.3 TB/s).
// ---------------------------------------------------------------------------

typedef float v2f __attribute__((ext_vector_type(2)));
typedef float v8f __attribute__((ext_vector_type(8)));

#define BD 8
#define ND 2048
#define IDm 64
#define HD 128
#define OD 8
#define LDm 3
#define KD 24

__device__ __forceinline__ v8f v8zero() {
  v8f z = {0.f, 0.f, 0.f, 0.f, 0.f, 0.f, 0.f, 0.f};
  return z;
}

// D(16x16,f32) = A(16x4,f32) x B(4x16,f32) + C
// Layout assumption (ISA 7.12.2): lane L: m/n = L%16, g = L/16.
//   A: a.x=A[m][2g], a.y=A[m][2g+1];  B: b.x=B[2g][n], b.y=B[2g+1][n]
//   C/D: c[r] = D[r + 8g][n]
__device__ __forceinline__ v8f wmma4(v2f a, v2f b, v8f c) {
  return __builtin_amdgcn_wmma_f32_16x16x4_f32(false, a, false, b, (short)0, c,
                                               false, false);
}

__device__ __forceinline__ float gelu_f(float x) {
  return 0.5f * x * (1.0f + erff(x * 0.70710678118654752f));
}
__device__ __forceinline__ float wave_sum32(float v) {
#pragma unroll
  for (int off = 16; off > 0; off >>= 1) v += __shfl_xor(v, off, 32);
  return v;
}
__device__ __forceinline__ void cswap(float& a, float& b) {
  float lo = fminf(a, b), hi = fmaxf(a, b);
  a = lo;
  b = hi;
}

// ---------------------------------------------------------------------------
// Kernel 1: encoder.  h = gelu(ln(x@W1.T+b1)); spatial = ln(h@W2.T+b2).
// Also writes sq[row] = ||x_row||^2 for the distance kernel.
// grid = B*N/16 = 1024, block = 256 (8 waves; wave w owns output cols 16w..).
// ---------------------------------------------------------------------------
__global__ void encoder_kernel(const float* __restrict__ x,
                               const float* __restrict__ w1,
                               const float* __restrict__ b1,
                               const float* __restrict__ g1,
                               const float* __restrict__ bb1,
                               const float* __restrict__ w2,
                               const float* __restrict__ b2,
                               const float* __restrict__ g2,
                               const float* __restrict__ bb2,
                               float* __restrict__ spatial,
                               float* __restrict__ sq) {
  __shared__ float xs[16 * IDm];  // 16x64 input tile
  __shared__ float hb[16 * HD];   // 16x128 work tile
  const int tid = threadIdx.x;
  const int wave = tid >> 5, lane = tid & 31;
  const int nl = lane & 15, g = lane >> 4;
  const int row0 = blockIdx.x * 16;

  ((float4*)xs)[tid] = ((const float4*)(x + (size_t)row0 * IDm))[tid];
  __syncthreads();

  if (tid < 16) {
    float s = 0.f;
#pragma unroll
    for (int i = 0; i < IDm; ++i) {
      float v = xs[tid * IDm + i];
      s += v * v;
    }
    sq[row0 + tid] = s;
  }

  const int col = wave * 16 + nl;
  // GEMM1: K=64 -> 16 WMMA steps
  v8f acc = v8zero();
#pragma unroll
  for (int kk = 0; kk < 16; ++kk) {
    const int k0 = kk * 4 + g * 2;
    v2f a;
    a.x = xs[nl * IDm + k0];
    a.y = xs[nl * IDm + k0 + 1];
    v2f bf;
    bf.x = w1[col * IDm + k0];
    bf.y = w1[col * IDm + k0 + 1];
    acc = wmma4(a, bf, acc);
  }
  {
    const float bias = b1[col];
#pragma unroll
    for (int r = 0; r < 8; ++r) hb[(r + 8 * g) * HD + col] = acc[r] + bias;
  }
  __syncthreads();

  // LN + GELU per row; wave handles rows 2w, 2w+1; lane owns 4 cols.
#pragma unroll
  for (int rr = 0; rr < 2; ++rr) {
    const int row = wave * 2 + rr;
    float v[4], s = 0.f, s2 = 0.f;
#pragma unroll
    for (int j = 0; j < 4; ++j) {
      v[j] = hb[row * HD + lane * 4 + j];
      s += v[j];
      s2 += v[j] * v[j];
    }
    s = wave_sum32(s);
    s2 = wave_sum32(s2);
    const float mean = s * (1.f / HD);
    const float var = s2 * (1.f / HD) - mean * mean;
    const float rstd = rsqrtf(var + 1e-5f);
#pragma unroll
    for (int j = 0; j < 4; ++j) {
      const int c = lane * 4 + j;
      float t = (v[j] - mean) * rstd * g1[c] + bb1[c];
      hb[row * HD + c] = gelu_f(t);
    }
  }
  __syncthreads();

  // GEMM2: K=128 -> 32 WMMA steps
  v8f acc2 = v8zero();
#pragma unroll
  for (int kk = 0; kk < 32; ++kk) {
    const int k0 = kk * 4 + g * 2;
    v2f a;
    a.x = hb[nl * HD + k0];
    a.y = hb[nl * HD + k0 + 1];
    v2f bf;
    bf.x = w2[col * HD + k0];
    bf.y = w2[col * HD + k0 + 1];
    acc2 = wmma4(a, bf, acc2);
  }
  __syncthreads();  // all waves done reading hb as A
  {
    const float bias = b2[col];
#pragma unroll
    for (int r = 0; r < 8; ++r) hb[(r + 8 * g) * HD + col] = acc2[r] + bias;
  }
  __syncthreads();

#pragma unroll
  for (int rr = 0; rr < 2; ++rr) {
    const int row = wave * 2 + rr;
    float v[4], s = 0.f, s2 = 0.f;
#pragma unroll
    for (int j = 0; j < 4; ++j) {
      v[j] = hb[row * HD + lane * 4 + j];
      s += v[j];
      s2 += v[j] * v[j];
    }
    s = wave_sum32(s);
    s2 = wave_sum32(s2);
    const float mean = s * (1.f / HD);
    const float var = s2 * (1.f / HD) - mean * mean;
    const float rstd = rsqrtf(var + 1e-5f);
#pragma unroll
    for (int j = 0; j < 4; ++j) {
      const int c = lane * 4 + j;
      spatial[(size_t)(row0 + row) * HD + c] =
          (v[j] - mean) * rstd * g2[c] + bb2[c];
    }
  }
}

// ---------------------------------------------------------------------------
// Kernel 2: density (dominant FLOPs). For a 16-query tile, stream all 2048
// candidates of the batch through WMMA Gram tiles (queries = N dim so each
// lane owns one query's top-6 list), merge per-lane lists in LDS.
// grid = 1024, block = 256.
// ---------------------------------------------------------------------------
__global__ void density_kernel(const float* __restrict__ x,
                               const float* __restrict__ sq,
                               float* __restrict__ dens) {
  __shared__ float xq[16 * IDm];  // 4 KB query tile
  __shared__ float sqb[ND];       // 8 KB batch norms
  __shared__ float mrg[16 * 96];  // 6 KB merge buffer
  const int tid = threadIdx.x, wave = tid >> 5, lane = tid & 31;
  const int nl = lane & 15, g = lane >> 4;
  const int q0 = blockIdx.x * 16;
  const int b = q0 >> 11;
  const int base = b << 11;

  ((float4*)xq)[tid] = ((const float4*)(x + (size_t)q0 * IDm))[tid];
  for (int i = tid; i < ND; i += 256) sqb[i] = sq[base + i];
  __syncthreads();

  // Query B-fragments held for the whole candidate loop.
  v2f bq[16];
#pragma unroll
  for (int kk = 0; kk < 16; ++kk) {
    const int k0 = kk * 4 + g * 2;
    bq[kk].x = xq[nl * IDm + k0];
    bq[kk].y = xq[nl * IDm + k0 + 1];
  }
  const float sqq = sqb[(q0 - base) + nl];
  float t0 = 3e38f, t1 = 3e38f, t2 = 3e38f, t3 = 3e38f, t4 = 3e38f, t5 = 3e38f;

  for (int c = wave; c < 128; c += 8) {  // wave-strided candidate chunks
    const int m0 = c * 16;
    const float* xa = x + (size_t)(base + m0) * IDm;
    v8f acc = v8zero();
#pragma unroll
    for (int kk = 0; kk < 16; ++kk) {
      const int k0 = kk * 4 + g * 2;
      v2f a;
      a.x = xa[nl * IDm + k0];
      a.y = xa[nl * IDm + k0 + 1];
      acc = wmma4(a, bq[kk], acc);
    }
#pragma unroll
    for (int r = 0; r < 8; ++r) {
      const int ml = m0 + r + 8 * g;
      const float d2 = sqb[ml] + sqq - 2.f * acc[r];
      const float d = sqrtf(fmaxf(d2, 0.f));
      t5 = fminf(t5, d);  // evict largest if d enters
      cswap(t4, t5);
      cswap(t3, t4);
      cswap(t2, t3);
      cswap(t1, t2);
      cswap(t0, t1);
    }
  }
  {
    const int list = wave * 2 + g;
    float* m = &mrg[nl * 96 + list * 6];
    m[0] = t0; m[1] = t1; m[2] = t2; m[3] = t3; m[4] = t4; m[5] = t5;
  }
  __syncthreads();
  if (tid < 16) {
    float f0 = 3e38f, f1 = 3e38f, f2 = 3e38f, f3 = 3e38f, f4 = 3e38f,
          f5 = 3e38f;
    for (int j = 0; j < 96; ++j) {
      const float d = mrg[tid * 96 + j];
      f5 = fminf(f5, d);
      cswap(f4, f5);
      cswap(f3, f4);
      cswap(f2, f3);
      cswap(f1, f2);
      cswap(f0, f1);
    }
    // drop self (smallest ~0), mean of next 5, tanh
    dens[q0 + tid] = tanhf((f1 + f2 + f3 + f4 + f5) * 0.2f);
  }
}

// ---------------------------------------------------------------------------
// Kernel 3: spatial head.  s = gelu(ln(concat(spatial,density)@W.T+b));
// pred_scale = softplus(s@w2.T+b2); pred_time = 5 - 1.5*dens + 0.5*ps.
// grid = 1024, block = 256.  K=128 on WMMA + rank-1 VALU for column 128.
// ---------------------------------------------------------------------------
__global__ void sphead_kernel(const float* __restrict__ spatial,
                              const float* __restrict__ dens,
                              const float* __restrict__ w1,  // (128,129)
                              const float* __restrict__ b1,
                              const float* __restrict__ gg,
                              const float* __restrict__ bb,
                              const float* __restrict__ w2,  // (1,128)
                              const float* __restrict__ b2,
                              float* __restrict__ pt) {
  __shared__ float sp[16 * HD];
  __shared__ float sb[16 * HD];
  __shared__ float dl[16];
  const int tid = threadIdx.x, wave = tid >> 5, lane = tid & 31;
  const int nl = lane & 15, g = lane >> 4;
  const int row0 = blockIdx.x * 16;

  ((float4*)sp)[tid] = ((const float4*)(spatial + (size_t)row0 * HD))[tid];
  ((float4*)sp)[tid + 256] =
      ((const float4*)(spatial + (size_t)row0 * HD))[tid + 256];
  if (tid < 16) dl[tid] = dens[row0 + tid];
  __syncthreads();

  const int col = wave * 16 + nl;
  v8f acc = v8zero();
#pragma unroll
  for (int kk = 0; kk < 32; ++kk) {
    const int k0 = kk * 4 + g * 2;
    v2f a;
    a.x = sp[nl * HD + k0];
    a.y = sp[nl * HD + k0 + 1];
    v2f bf;
    bf.x = w1[col * 129 + k0];
    bf.y = w1[col * 129 + k0 + 1];
    acc = wmma4(a, bf, acc);
  }
  {
    const float bias = b1[col], wlast = w1[col * 129 + 128];
#pragma unroll
    for (int r = 0; r < 8; ++r)
      sb[(r + 8 * g) * HD + col] = acc[r] + bias + dl[r + 8 * g] * wlast;
  }
  __syncthreads();

#pragma unroll
  for (int rr = 0; rr < 2; ++rr) {
    const int row = wave * 2 + rr;
    float v[4], s = 0.f, s2 = 0.f;
#pragma unroll
    for (int j = 0; j < 4; ++j) {
      v[j] = sb[row * HD + lane * 4 + j];
      s += v[j];
      s2 += v[j] * v[j];
    }
    s = wave_sum32(s);
    s2 = wave_sum32(s2);
    const float mean = s * (1.f / HD);
    const float var = s2 * (1.f / HD) - mean * mean;
    const float rstd = rsqrtf(var + 1e-5f);
    float part = 0.f;
#pragma unroll
    for (int j = 0; j < 4; ++j) {
      const int c = lane * 4 + j;
      const float t = gelu_f((v[j] - mean) * rstd * gg[c] + bb[c]);
      part += t * w2[c];
    }
    part = wave_sum32(part);
    if (lane == 0) {
      const float ps = part + b2[0];
      const float spl = (ps > 20.f) ? ps : log1pf(expf(ps));
      pt[row0 + row] = 5.f - 1.5f * dl[row] + 0.5f * spl;
    }
  }
}

// ---------------------------------------------------------------------------
// slot_state[b,o,:] = object_space[o,:]   (B*O*H = 8192)
// ---------------------------------------------------------------------------
__global__ void init_slots(const float* __restrict__ objsp,
                           float* __restrict__ slot) {
  const int idx = blockIdx.x * 256 + threadIdx.x;
  if (idx < BD * OD * HD) slot[idx] = objsp[idx & (OD * HD - 1)];
}

// ---------------------------------------------------------------------------
// Attention kernel: per (b,n) thread computes 24 logits (dr2 collapses to 8
// object vectors), softmax over k, writes attn; deterministic per-block
// partial sums (fixed-order reduction, no float atomics).
// grid = B*N/256 = 64, block = 256.
// ---------------------------------------------------------------------------
__global__ void attn_kernel(const float* __restrict__ spatial,
                            const float* __restrict__ pt,
                            const float* __restrict__ dens,
                            const float* __restrict__ slot,
                            const float* __restrict__ ltimes,
                            const float* __restrict__ bhz,
                            const float* __restrict__ hscale,
                            float* __restrict__ attn,
                            float* __restrict__ partial) {
  __shared__ float ob[OD * HD];  // 8 object vectors
  __shared__ float psum[8][KD];
  const int tid = threadIdx.x, wave = tid >> 5, lane = tid & 31;
  const int row = blockIdx.x * 256 + tid;
  const int b = row >> 11, n = row & 2047;

  ((float4*)ob)[tid] = ((const float4*)(slot + (size_t)b * OD * HD))[tid];
  __syncthreads();

  float dr2o[OD];
#pragma unroll
  for (int o = 0; o < OD; ++o) dr2o[o] = 0.f;
  const float4* fv4 = (const float4*)(spatial + (size_t)row * HD);
  for (int i4 = 0; i4 < 32; ++i4) {
    const float4 f = fv4[i4];
#pragma unroll
    for (int o = 0; o < OD; ++o) {
      const float4 s = ((const float4*)(ob + o * HD))[i4];
      const float dx = f.x - s.x, dy = f.y - s.y, dz = f.z - s.z,
                  dw = f.w - s.w;
      dr2o[o] += dx * dx + dy * dy + dz * dz + dw * dw;
    }
  }
  const float p = pt[row], dv = dens[row], hs = hscale[0];
  const float lt[3] = {ltimes[0], ltimes[1], ltimes[2]};
  const float bh[3] = {bhz[0], bhz[1], bhz[2]};
  float lg[KD], mx = -3e38f;
#pragma unroll
  for (int k = 0; k < KD; ++k) {
    const int o = k / 3, l = k % 3;
    const float dt = p - lt[l];
    const float dr2 = dr2o[o];
    const float ls = dt * dt - dr2;
    const float sd = sqrtf(fabsf(ls) + 1e-6f);  // |d_L|
    const float r = sqrtf(dr2);
    const float adt = fabsf(dt);
    float hz = bh[l] + hs * (dv - 0.5f);
    hz = fminf(fmaxf(hz, 0.1f), 1.0f);
    const float cone =
        hz - r / (adt + 1e-6f) - 10.f * fmaxf(-dt, 0.f) - 5.f * fmaxf(r - adt, 0.f);
    lg[k] = -sd + 0.5f * tanhf(cone);
    mx = fmaxf(mx, lg[k]);
  }
  float se = 0.f;
#pragma unroll
  for (int k = 0; k < KD; ++k) {
    lg[k] = expf((lg[k] - mx) * 10.f);  // /TAU
    se += lg[k];
  }
  const float inv = 1.f / se;
#pragma unroll
  for (int k = 0; k < KD; ++k) {
    const float a = lg[k] * inv;
    attn[((size_t)(b * KD + k) << 11) + n] = a;
    const float ws = wave_sum32(a);
    if (lane == 0) psum[wave][k] = ws;
  }
  __syncthreads();
  if (tid < KD) {
    float s = 0.f;
#pragma unroll
    for (int w = 0; w < 8; ++w) s += psum[w][tid];
    partial[blockIdx.x * KD + tid] = s;
  }
}

// ---------------------------------------------------------------------------
// upd[b,k,:] = sum_n (attn/(sum_n attn + 1e-8)) * spatial[n,:]
// grid = B = 8, block = 256 (thread owns i = tid&127, 12 consecutive k's).
// ---------------------------------------------------------------------------
__global__ void upd_kernel(const float* __restrict__ spatial,
                           const float* __restrict__ attn,
                           const float* __restrict__ partial,
                           float* __restrict__ upd) {
  __shared__ float wl[KD * 128];
  __shared__ float winv[KD];
  const int tid = threadIdx.x, b = blockIdx.x;
  const int i = tid & 127, kb = (tid >> 7) * 12;
  if (tid < KD) {
    float s = 0.f;
#pragma unroll
    for (int blk = 0; blk < 8; ++blk) s += partial[(b * 8 + blk) * KD + tid];
    winv[tid] = 1.f / (s + 1e-8f);
  }
  float a2[12];
#pragma unroll
  for (int j = 0; j < 12; ++j) a2[j] = 0.f;
  for (int c = 0; c < 16; ++c) {  // n chunks of 128
    __syncthreads();
#pragma unroll
    for (int j = 0; j < 12; ++j) {
      const int idx = tid * 12 + j;  // k*128 + n_local
      const int kk = idx >> 7, nn = idx & 127;
      wl[idx] = attn[((size_t)(b * KD + kk) << 11) + c * 128 + nn] * winv[kk];
    }
    __syncthreads();
    for (int nn = 0; nn < 128; ++nn) {
      const float sv = spatial[((size_t)(b * 2048 + c * 128 + nn)) * HD + i];
#pragma unroll
      for (int j = 0; j < 12; ++j) a2[j] += wl[(kb + j) * 128 + nn] * sv;
    }
  }
#pragma unroll
  for (int j = 0; j < 12; ++j)
    upd[((size_t)(b * KD + kb + j)) * HD + i] = a2[j];
}

// ---------------------------------------------------------------------------
// GRU + MLP per (b,o).  NOTE the reference quirk: old_flat is the ORIGINAL
// object_space on every iteration (never updated). Writes new slot spatial
// state and the slots output (times column + repeated object vectors).
// grid = B*O = 64, block = 128.
// ---------------------------------------------------------------------------
__global__ void gru_kernel(const float* __restrict__ upd,
                           const float* __restrict__ objsp,
                           const float* __restrict__ wih,
                           const float* __restrict__ whh,
                           const float* __restrict__ bih,
                           const float* __restrict__ bhh,
                           const float* __restrict__ mw1,
                           const float* __restrict__ mb1,
                           const float* __restrict__ mw2,
                           const float* __restrict__ mb2,
                           const float* __restrict__ ngm,
                           const float* __restrict__ nbm,
                           const float* __restrict__ ltimes,
                           float* __restrict__ slot,
                           float* __restrict__ slots_out) {
  __shared__ float u[HD], oldv[HD], mid[HD], red[8];
  const int bo = blockIdx.x, b = bo >> 3, o = bo & 7, h = threadIdx.x;
  const size_t kbase = (size_t)(b * KD + o * 3);
  u[h] = upd[(kbase + 0) * HD + h] + upd[(kbase + 1) * HD + h] +
         upd[(kbase + 2) * HD + h];
  oldv[h] = objsp[o * HD + h];
  __syncthreads();

  float gi_r = bih[h], gi_z = bih[HD + h], gi_n = bih[2 * HD + h];
  float gh_r = bhh[h], gh_z = bhh[HD + h], gh_n = bhh[2 * HD + h];
  for (int i = 0; i < HD; ++i) {
    const float uv = u[i], ov = oldv[i];
    gi_r += uv * wih[h * HD + i];
    gi_z += uv * wih[(HD + h) * HD + i];
    gi_n += uv * wih[(2 * HD + h) * HD + i];
    gh_r += ov * whh[h * HD + i];
    gh_z += ov * whh[(HD + h) * HD + i];
    gh_n += ov * whh[(2 * HD + h) * HD + i];
  }
  const float rg = 1.f / (1.f + expf(-(gi_r + gh_r)));
  const float zg = 1.f / (1.f + expf(-(gi_z + gh_z)));
  const float ngv = tanhf(gi_n + rg * gh_n);
  const float newv = (1.f - zg) * ngv + zg * oldv[h];
  __syncthreads();  // done reading u/oldv

  // LayerNorm(new) across 128 via wave shfl + 4-partial LDS combine
  float s = wave_sum32(newv), s2 = wave_sum32(newv * newv);
  const int wv = h >> 5;
  if ((h & 31) == 0) {
    red[wv] = s;
    red[4 + wv] = s2;
  }
  __syncthreads();
  s = red[0] + red[1] + red[2] + red[3];
  s2 = red[4] + red[5] + red[6] + red[7];
  const float mean = s * (1.f / HD), var = s2 * (1.f / HD) - mean * mean;
  const float rstd = rsqrtf(var + 1e-5f);
  u[h] = (newv - mean) * rstd * ngm[h] + nbm[h];
  __syncthreads();

  float m1 = mb1[h];
  for (int i = 0; i < HD; ++i) m1 += u[i] * mw1[h * HD + i];
  mid[h] = gelu_f(m1);
  __syncthreads();
  float t2 = mb2[h];
  for (int i = 0; i < HD; ++i) t2 += mid[i] * mw2[h * HD + i];
  const float outv = newv + 0.2f * t2;

  slot[(size_t)bo * HD + h] = outv;
#pragma unroll
  for (int l = 0; l < 3; ++l) slots_out[(kbase + l) * 129 + 1 + h] = outv;
  if (h < 3) slots_out[(kbase + h) * 129] = ltimes[h];
}

// ---------------------------------------------------------------------------
extern "C" void kernel_launch(void* const* d_in, const int* in_sizes, int n_in,
                              void* d_out, int out_size, void* d_ws,
                              size_t ws_size, hipStream_t stream) {
  (void)in_sizes;
  (void)n_in;
  (void)out_size;
  (void)ws_size;
  const float* x = (const float*)d_in[0];
  const float* enc_w1 = (const float*)d_in[1];
  const float* enc_b1 = (const float*)d_in[2];
  const float* enc_g1 = (const float*)d_in[3];
  const float* enc_bb1 = (const float*)d_in[4];
  const float* enc_w2 = (const float*)d_in[5];
  const float* enc_b2 = (const float*)d_in[6];
  const float* enc_g2 = (const float*)d_in[7];
  const float* enc_bb2 = (const float*)d_in[8];
  const float* sp_w1 = (const float*)d_in[9];
  const float* sp_b1 = (const float*)d_in[10];
  const float* sp_g = (const float*)d_in[11];
  const float* sp_bb = (const float*)d_in[12];
  const float* sp_w2 = (const float*)d_in[13];
  const float* sp_b2 = (const float*)d_in[14];
  const float* object_space = (const float*)d_in[15];
  const float* horizon_scale = (const float*)d_in[16];
  const float* level_times = (const float*)d_in[17];
  const float* base_horizons = (const float*)d_in[18];
  const float* gru_w_ih = (const float*)d_in[19];
  const float* gru_w_hh = (const float*)d_in[20];
  const float* gru_b_ih = (const float*)d_in[21];
  const float* gru_b_hh = (const float*)d_in[22];
  const float* mlp_w1 = (const float*)d_in[23];
  const float* mlp_b1 = (const float*)d_in[24];
  const float* mlp_w2 = (const float*)d_in[25];
  const float* mlp_b2 = (const float*)d_in[26];
  const float* norm_g = (const float*)d_in[27];
  const float* norm_b = (const float*)d_in[28];

  float* out = (float*)d_out;
  float* out_slots = out;                      // B*K*129 = 24768
  float* out_attn = out + 24768;               // B*K*N  = 393216
  float* out_pt = out + 24768 + 393216;        // B*N    = 16384
  float* out_dens = out + 24768 + 393216 + 16384;  // B*N = 16384

  float* w = (float*)d_ws;                 // ~8.6 MB of f32 scratch
  float* spatial = w;                      // B*N*H   = 2097152
  float* sq = spatial + (size_t)BD * ND * HD;   // B*N = 16384
  float* slot_state = sq + BD * ND;        // B*O*H   = 8192
  float* partial = slot_state + BD * OD * HD;   // 64*24 = 1536
  float* updbuf = partial + 64 * KD;       // B*K*H   = 24576

  encoder_kernel<<<BD * ND / 16, 256, 0, stream>>>(
      x, enc_w1, enc_b1, enc_g1, enc_bb1, enc_w2, enc_b2, enc_g2, enc_bb2,
      spatial, sq);
  density_kernel<<<BD * ND / 16, 256, 0, stream>>>(x, sq, out_dens);
  sphead_kernel<<<BD * ND / 16, 256, 0, stream>>>(
      spatial, out_dens, sp_w1, sp_b1, sp_g, sp_bb, sp_w2, sp_b2, out_pt);
  init_slots<<<32, 256, 0, stream>>>(object_space, slot_state);

  for (int it = 0; it < 3; ++it) {
    attn_kernel<<<BD * ND / 256, 256, 0, stream>>>(
        spatial, out_pt, out_dens, slot_state, level_times, base_horizons,
        horizon_scale, out_attn, partial);
    upd_kernel<<<BD, 256, 0, stream>>>(spatial, out_attn, partial, updbuf);
    gru_kernel<<<BD * OD, 128, 0, stream>>>(
        updbuf, object_space, gru_w_ih, gru_w_hh, gru_b_ih, gru_b_hh, mlp_w1,
        mlp_b1, mlp_w2, mlp_b2, norm_g, norm_b, level_times, slot_state,
        out_slots);
  }
}